// HCFCSAGE_34763465294563
// MI455X (gfx1250) — compile-verified
//
#include <hip/hip_runtime.h>
#include <hip/hip_bf16.h>
#include <cstddef>

#define IN_DIM 12
#define HID 128
#define OUTC 13
#define OUTP 16   // padded out dim for layer 2

typedef __attribute__((ext_vector_type(2))) float v2f;
typedef __attribute__((ext_vector_type(8))) float v8f;

// ---------------------------------------------------------------- utilities

__global__ void zero_f32(float* __restrict__ p, int n) {
    int i = blockIdx.x * blockDim.x + threadIdx.x;
    if (i < n) p[i] = 0.0f;
}

__global__ void count_deg(const int* __restrict__ dst, float* __restrict__ deg, int E) {
    int e = blockIdx.x * blockDim.x + threadIdx.x;
    if (e < E) unsafeAtomicAdd(&deg[dst[e]], 1.0f);
}

__global__ void inv_deg_inplace(float* __restrict__ d, int n) {
    int i = blockIdx.x * blockDim.x + threadIdx.x;
    if (i < n) d[i] = 1.0f / fmaxf(d[i], 1.0f);
}

// Wt[n*din + k] = W[k*dout + n], rows n >= dout zero-padded
__global__ void transpose_pad(const float* __restrict__ W, float* __restrict__ Wt,
                              int din, int dout, int dout_pad) {
    int idx = blockIdx.x * blockDim.x + threadIdx.x;
    int total = din * dout_pad;
    if (idx >= total) return;
    int n = idx / din;
    int k = idx - n * din;
    Wt[idx] = (n < dout) ? W[(size_t)k * dout + n] : 0.0f;
}

// -------------------------------------------------------------- aggregation

// generic: one thread per (edge, feature); used for din=12
__global__ void aggregate_gen(const float* __restrict__ h, const int* __restrict__ src,
                              const int* __restrict__ dst, float* __restrict__ agg,
                              int E, int din) {
    int idx = blockIdx.x * blockDim.x + threadIdx.x;
    int total = E * din;
    if (idx >= total) return;
    int e = idx / din;
    int f = idx - e * din;
    unsafeAtomicAdd(&agg[(size_t)dst[e] * din + f], h[(size_t)src[e] * din + f]);
}

// din=128: one wave per edge, float4 per lane (coalesced b128 gather, 4 HW f32 atomics)
__global__ void aggregate_128(const float* __restrict__ h, const int* __restrict__ src,
                              const int* __restrict__ dst, float* __restrict__ agg, int E) {
    int e = blockIdx.x * (blockDim.x >> 5) + (threadIdx.x >> 5);
    if (e >= E) return;
    int lane = threadIdx.x & 31;
    int s = src[e], d = dst[e];
    const float4 v = *(const float4*)(h + (size_t)s * 128 + lane * 4);
    float* base = agg + (size_t)d * 128 + lane * 4;
    unsafeAtomicAdd(base + 0, v.x);
    unsafeAtomicAdd(base + 1, v.y);
    unsafeAtomicAdd(base + 2, v.z);
    unsafeAtomicAdd(base + 3, v.w);
}

// ------------------------------------------------------- fused SAGE + WMMA

// out[N][dout_pad] = act( (agg*invdeg) @ Wl + h @ Wr + b )
// One wave computes a 16x16 tile via V_WMMA_F32_16X16X4_F32.
// Wlt/Wrt are transposed: Wt[n*DIN + k].
template <int DIN>
__global__ void sage_gemm_wmma(const float* __restrict__ agg,
                               const float* __restrict__ h,
                               const float* __restrict__ invdeg,
                               const float* __restrict__ Wlt,
                               const float* __restrict__ Wrt,
                               const float* __restrict__ bias,
                               float* __restrict__ out,
                               int N, int dout, int dout_pad, int do_relu) {
    const int wavesPerBlock = blockDim.x >> 5;
    const int wid = blockIdx.x * wavesPerBlock + (threadIdx.x >> 5);
    const int nRowTiles = (N + 15) >> 4;
    const int nColTiles = dout_pad >> 4;
    if (wid >= nRowTiles * nColTiles) return;     // wave-uniform: EXEC stays full
    const int rowTile = wid / nColTiles;          // adjacent waves share a row tile (A reuse)
    const int colTile = wid - rowTile * nColTiles;

    const int m0 = rowTile << 4;
    const int n0 = colTile << 4;
    const int lane = threadIdx.x & 31;
    const int r = lane & 15;
    const int half = lane >> 4;
    const int koff = half * 2;                     // lanes 0-15: K=k0,k0+1; 16-31: K=k0+2,k0+3

    int rowA = m0 + r;
    if (rowA >= N) rowA = N - 1;                   // clamp loads; stores are guarded
    const float idg = invdeg[rowA];
    const int col = n0 + r;

    const float* aMean = agg + (size_t)rowA * DIN + koff;
    const float* aSelf = h   + (size_t)rowA * DIN + koff;
    const float* bL    = Wlt + (size_t)col  * DIN + koff;
    const float* bR    = Wrt + (size_t)col  * DIN + koff;

    v8f acc = {};
#pragma unroll
    for (int k0 = 0; k0 < DIN; k0 += 4) {
        v2f a = *(const v2f*)(aMean + k0);
        a *= idg;                                   // fused mean = sums * inv_deg
        v2f b = *(const v2f*)(bL + k0);
        acc = __builtin_amdgcn_wmma_f32_16x16x4_f32(false, a, false, b,
                                                    (short)0, acc, false, false);
    }
#pragma unroll
    for (int k0 = 0; k0 < DIN; k0 += 4) {
        v2f a = *(const v2f*)(aSelf + k0);
        v2f b = *(const v2f*)(bR + k0);
        acc = __builtin_amdgcn_wmma_f32_16x16x4_f32(false, a, false, b,
                                                    (short)0, acc, false, false);
    }

    if (col < dout) {
        const float bs = bias[col];
        float* outBase = out + (size_t)(m0 + 8 * half) * dout_pad + col;
        if (m0 + 16 <= N) {
            // full tile (always true when N % 16 == 0): 8 unguarded coalesced stores
#pragma unroll
            for (int v = 0; v < 8; ++v) {           // C/D layout: M = v + 8*half, N = r
                float val = acc[v] + bs;
                if (do_relu) val = fmaxf(val, 0.0f);
                outBase[(size_t)v * dout_pad] = val;
            }
        } else {
#pragma unroll
            for (int v = 0; v < 8; ++v) {
                int row = m0 + v + 8 * half;
                if (row < N) {
                    float val = acc[v] + bs;
                    if (do_relu) val = fmaxf(val, 0.0f);
                    out[(size_t)row * dout_pad + col] = val;
                }
            }
        }
    }
}

// ------------------------------------------------- sigmoid + hierarchy max

__global__ void hier_out(const float* __restrict__ logits, const int* __restrict__ R,
                         float* __restrict__ out, int N) {
    __shared__ int Rs[OUTC * OUTC];
    if (threadIdx.x < OUTC * OUTC) Rs[threadIdx.x] = R[threadIdx.x];
    __syncthreads();
    int n = blockIdx.x * blockDim.x + threadIdx.x;
    if (n >= N) return;
    float s[OUTC];
#pragma unroll
    for (int j = 0; j < OUTC; ++j) {
        float x = logits[(size_t)n * OUTP + j];
        s[j] = 1.0f / (1.0f + expf(-x));
    }
#pragma unroll
    for (int i = 0; i < OUTC; ++i) {
        float m = 0.0f;
#pragma unroll
        for (int j = 0; j < OUTC; ++j)
            if (Rs[i * OUTC + j]) m = fmaxf(m, s[j]);
        out[(size_t)n * OUTC + i] = m;
    }
}

// ------------------------------------------------------------------ driver

extern "C" void kernel_launch(void* const* d_in, const int* in_sizes, int n_in,
                              void* d_out, int out_size, void* d_ws, size_t ws_size,
                              hipStream_t stream) {
    const float* x   = (const float*)d_in[0];
    const float* Wl0 = (const float*)d_in[1];
    const float* Wr0 = (const float*)d_in[2];
    const float* b0  = (const float*)d_in[3];
    const float* Wl1 = (const float*)d_in[4];
    const float* Wr1 = (const float*)d_in[5];
    const float* b1  = (const float*)d_in[6];
    const float* Wl2 = (const float*)d_in[7];
    const float* Wr2 = (const float*)d_in[8];
    const float* b2  = (const float*)d_in[9];
    const int* edge  = (const int*)d_in[10];
    const int* R     = (const int*)d_in[11];
    float* out       = (float*)d_out;

    const int N = in_sizes[0] / IN_DIM;
    const int E = in_sizes[10] / 2;
    const int* src = edge;
    const int* dst = edge + E;

    // workspace layout
    float* invd   = (float*)d_ws;                    // N
    float* agg    = invd + N;                        // N*128 (layer0 uses first N*12)
    float* h1     = agg + (size_t)N * HID;           // N*128
    float* h2     = h1 + (size_t)N * HID;            // N*128
    float* logits = h2 + (size_t)N * HID;            // N*16
    float* Wl0t   = logits + (size_t)N * OUTP;       // 128*12
    float* Wr0t   = Wl0t + HID * IN_DIM;
    float* Wl1t   = Wr0t + HID * IN_DIM;             // 128*128
    float* Wr1t   = Wl1t + HID * HID;
    float* Wl2t   = Wr1t + HID * HID;                // 16*128
    float* Wr2t   = Wl2t + OUTP * HID;

    const int B = 256;
    const int wavesPerBlock = B >> 5;

    // degrees -> inverse degrees (shared by all layers)
    zero_f32<<<(N + B - 1) / B, B, 0, stream>>>(invd, N);
    count_deg<<<(E + B - 1) / B, B, 0, stream>>>(dst, invd, E);
    inv_deg_inplace<<<(N + B - 1) / B, B, 0, stream>>>(invd, N);

    // transposed (and padded) weights
    transpose_pad<<<(IN_DIM * HID + B - 1) / B, B, 0, stream>>>(Wl0, Wl0t, IN_DIM, HID, HID);
    transpose_pad<<<(IN_DIM * HID + B - 1) / B, B, 0, stream>>>(Wr0, Wr0t, IN_DIM, HID, HID);
    transpose_pad<<<(HID * HID + B - 1) / B, B, 0, stream>>>(Wl1, Wl1t, HID, HID, HID);
    transpose_pad<<<(HID * HID + B - 1) / B, B, 0, stream>>>(Wr1, Wr1t, HID, HID, HID);
    transpose_pad<<<(HID * OUTP + B - 1) / B, B, 0, stream>>>(Wl2, Wl2t, HID, OUTC, OUTP);
    transpose_pad<<<(HID * OUTP + B - 1) / B, B, 0, stream>>>(Wr2, Wr2t, HID, OUTC, OUTP);

    // ---- layer 0: x(12) -> h1(128), relu
    zero_f32<<<(N * IN_DIM + B - 1) / B, B, 0, stream>>>(agg, N * IN_DIM);
    aggregate_gen<<<(E * IN_DIM + B - 1) / B, B, 0, stream>>>(x, src, dst, agg, E, IN_DIM);
    {
        int tiles = ((N + 15) / 16) * (HID / 16);
        sage_gemm_wmma<IN_DIM><<<(tiles + wavesPerBlock - 1) / wavesPerBlock, B, 0, stream>>>(
            agg, x, invd, Wl0t, Wr0t, b0, h1, N, HID, HID, 1);
    }

    // ---- layer 1: h1(128) -> h2(128), relu
    zero_f32<<<(N * HID + B - 1) / B, B, 0, stream>>>(agg, N * HID);
    aggregate_128<<<(E + wavesPerBlock - 1) / wavesPerBlock, B, 0, stream>>>(h1, src, dst, agg, E);
    {
        int tiles = ((N + 15) / 16) * (HID / 16);
        sage_gemm_wmma<HID><<<(tiles + wavesPerBlock - 1) / wavesPerBlock, B, 0, stream>>>(
            agg, h1, invd, Wl1t, Wr1t, b1, h2, N, HID, HID, 1);
    }

    // ---- layer 2: h2(128) -> logits(13, padded 16), no relu
    zero_f32<<<(N * HID + B - 1) / B, B, 0, stream>>>(agg, N * HID);
    aggregate_128<<<(E + wavesPerBlock - 1) / wavesPerBlock, B, 0, stream>>>(h2, src, dst, agg, E);
    {
        int tiles = ((N + 15) / 16) * (OUTP / 16);
        sage_gemm_wmma<HID><<<(tiles + wavesPerBlock - 1) / wavesPerBlock, B, 0, stream>>>(
            agg, h2, invd, Wl2t, Wr2t, b2, logits, N, OUTC, OUTP, 0);
    }

    // ---- sigmoid + hierarchical max readout
    hier_out<<<(N + B - 1) / B, B, 0, stream>>>(logits, R, out, N);
}